// GlobalBlock_19877108646540
// MI455X (gfx1250) — compile-verified
//
#include <hip/hip_runtime.h>

#define D      256
#define G      1024
#define DIN    768
#define DHID   512
#define DOUT   256
#define STRA   772   // 768 + 4 pad -> lane stride 4 banks, conflict-free
#define STRH   516   // 512 + 4 pad

typedef __attribute__((ext_vector_type(2))) float v2f;
typedef __attribute__((ext_vector_type(8))) float v8f;

// ---------------------------------------------------------------------------
// Segment sum over sorted ids. Block = 64 threads, each owns 4 columns
// (float4). Block processes `rows_per_block` contiguous rows, accumulating in
// registers while the (uniform, scalar-loaded) graph id is unchanged; flushes
// with f32 global atomics only at segment boundaries. Fully coalesced b128
// streaming reads -> HBM roofline.
// ---------------------------------------------------------------------------
__global__ __launch_bounds__(64) void seg_sum_kernel(
    const float* __restrict__ x, const int* __restrict__ ids,
    float* __restrict__ out, int n_rows, int rows_per_block) {
  const int c  = threadIdx.x * 4;
  const int r0 = blockIdx.x * rows_per_block;
  int rend = r0 + rows_per_block;
  if (rend > n_rows) rend = n_rows;

  float4 acc = make_float4(0.f, 0.f, 0.f, 0.f);
  int cur = ids[r0];                       // uniform -> s_load
  for (int r = r0; r < rend; ++r) {
    const int g = ids[r];                  // uniform -> s_load
    if (g != cur) {                        // uniform branch
      atomicAdd(out + cur * D + c + 0, acc.x);
      atomicAdd(out + cur * D + c + 1, acc.y);
      atomicAdd(out + cur * D + c + 2, acc.z);
      atomicAdd(out + cur * D + c + 3, acc.w);
      acc = make_float4(0.f, 0.f, 0.f, 0.f);
      cur = g;
    }
    const float4 v = *reinterpret_cast<const float4*>(x + (size_t)r * D + c);
    acc.x += v.x; acc.y += v.y; acc.z += v.z; acc.w += v.w;
  }
  atomicAdd(out + cur * D + c + 0, acc.x);
  atomicAdd(out + cur * D + c + 1, acc.y);
  atomicAdd(out + cur * D + c + 2, acc.z);
  atomicAdd(out + cur * D + c + 3, acc.w);
}

// ---------------------------------------------------------------------------
// Fused 2-layer MLP using V_WMMA_F32_16X16X4_F32 (full fp32 precision).
// Block = 256 threads (8 waves), handles 16 output rows.
//
// WMMA lane layouts (ISA 7.12.2, 32-bit):
//   A 16x4 : lane<16 -> M=lane,    vgpr0=K0, vgpr1=K1
//            lane>=16 -> M=lane-16, vgpr0=K2, vgpr1=K3
//   B 4x16 : vgpr v, lane<16 -> row K=v,   N=lane
//                    lane>=16 -> row K=v+2, N=lane-16
//   C/D    : vgpr i, lane<16 -> M=i,   N=lane
//                    lane>=16 -> M=i+8, N=lane-16
// ---------------------------------------------------------------------------
__global__ __launch_bounds__(256) void mlp_kernel(
    const float* __restrict__ edge_agg, const float* __restrict__ node_agg,
    const float* __restrict__ globals_,
    const float* __restrict__ W1, const float* __restrict__ b1,
    const float* __restrict__ W2, const float* __restrict__ b2,
    float* __restrict__ out) {
  __shared__ float Acol[16 * STRA];   // collected tile [16, 768], padded
  __shared__ float Hs[16 * STRH];     // hidden tile    [16, 512], padded

  const int tid  = threadIdx.x;
  const int lane = tid & 31;
  const int wave = tid >> 5;          // 0..7
  const int half = lane >> 4;         // 0: lanes 0-15, 1: lanes 16-31
  const int m    = lane & 15;
  const int row0 = blockIdx.x * 16;

  // Stage collected = concat(edge_agg, node_agg, globals) rows into LDS.
  for (int idx = tid; idx < 16 * DIN; idx += 256) {
    const int r = idx / DIN;
    const int cc = idx - r * DIN;
    const int g = row0 + r;
    float v;
    if (cc < D)          v = edge_agg[g * D + cc];
    else if (cc < 2 * D) v = node_agg[g * D + (cc - D)];
    else                 v = globals_[g * D + (cc - 2 * D)];
    Acol[r * STRA + cc] = v;
  }
  __syncthreads();

  // ---- Layer 1: h[16,512] = relu(collected @ W1 + b1); 32 N-tiles, 4/wave.
  for (int tt = 0; tt < 4; ++tt) {
    const int nb = (wave * 4 + tt) * 16;
    v8f acc = {};
    const float* ap = &Acol[m * STRA + half * 2];
    const float* bp = &W1[(half * 2) * DHID + nb + m];
    for (int k = 0; k < DIN; k += 4) {
      v2f a; a.x = ap[k];        a.y = ap[k + 1];
      v2f b; b.x = bp[k * DHID]; b.y = bp[k * DHID + DHID];
      acc = __builtin_amdgcn_wmma_f32_16x16x4_f32(
          false, a, false, b, (short)0, acc, false, false);
    }
    const float bias = b1[nb + m];
#pragma unroll
    for (int i = 0; i < 8; ++i) {
      const float h = acc[i] + bias;
      Hs[(i + half * 8) * STRH + nb + m] = h > 0.f ? h : 0.f;
    }
  }
  __syncthreads();

  // ---- Layer 2: out[16,256] = h @ W2 + b2; 16 N-tiles, 2/wave.
  for (int tt = 0; tt < 2; ++tt) {
    const int nb = (wave * 2 + tt) * 16;
    v8f acc = {};
    const float* ap = &Hs[m * STRH + half * 2];
    const float* bp = &W2[(half * 2) * DOUT + nb + m];
    for (int k = 0; k < DHID; k += 4) {
      v2f a; a.x = ap[k];        a.y = ap[k + 1];
      v2f b; b.x = bp[k * DOUT]; b.y = bp[k * DOUT + DOUT];
      acc = __builtin_amdgcn_wmma_f32_16x16x4_f32(
          false, a, false, b, (short)0, acc, false, false);
    }
    const float bias = b2[nb + m];
#pragma unroll
    for (int i = 0; i < 8; ++i) {
      out[(size_t)(row0 + i + half * 8) * DOUT + nb + m] = acc[i] + bias;
    }
  }
}

extern "C" void kernel_launch(void* const* d_in, const int* in_sizes, int n_in,
                              void* d_out, int out_size, void* d_ws, size_t ws_size,
                              hipStream_t stream) {
  const float* nodes    = (const float*)d_in[0];
  const float* edges    = (const float*)d_in[1];
  const float* globals_ = (const float*)d_in[2];
  const int*   node_ids = (const int*)d_in[3];
  const int*   edge_ids = (const int*)d_in[4];
  const float* W1       = (const float*)d_in[5];
  const float* b1       = (const float*)d_in[6];
  const float* W2       = (const float*)d_in[7];
  const float* b2       = (const float*)d_in[8];
  float* out = (float*)d_out;

  float* edge_agg = (float*)d_ws;          // [G, D]
  float* node_agg = edge_agg + (size_t)G * D;  // [G, D]

  hipMemsetAsync(d_ws, 0, (size_t)2 * G * D * sizeof(float), stream);

  const int n_nodes = in_sizes[0] / D;     // 65536
  const int n_edges = in_sizes[1] / D;     // 262144
  const int ROWS = 256;

  seg_sum_kernel<<<(n_edges + ROWS - 1) / ROWS, 64, 0, stream>>>(
      edges, edge_ids, edge_agg, n_edges, ROWS);
  seg_sum_kernel<<<(n_nodes + ROWS - 1) / ROWS, 64, 0, stream>>>(
      nodes, node_ids, node_agg, n_nodes, ROWS);

  mlp_kernel<<<G / 16, 256, 0, stream>>>(
      edge_agg, node_agg, globals_, W1, b1, W2, b2, out);
}